// GAT_31353261261175
// MI455X (gfx1250) — compile-verified
//
#include <hip/hip_runtime.h>
#include <hip/hip_bf16.h>
#include <math.h>

// ---------------------------------------------------------------------------
// CDNA5 (gfx1250) GAT forward: two GATConv layers + log_softmax.
// fp32 WMMA (V_WMMA_F32_16X16X4_F32) for the dense transforms, TDM
// (tensor_load_to_lds) for weight staging, bandwidth-oriented edge kernels
// (f32 global atomics) for the softmax-aggregation.
// ---------------------------------------------------------------------------

typedef __attribute__((ext_vector_type(2))) float        v2f;
typedef __attribute__((ext_vector_type(8))) float        v8f;
typedef __attribute__((ext_vector_type(4))) unsigned int v4u;
typedef __attribute__((ext_vector_type(4))) int          v4i;
typedef __attribute__((ext_vector_type(8))) int          v8i;

#if defined(__has_builtin)
#if __has_builtin(__builtin_amdgcn_tensor_load_to_lds) && \
    __has_builtin(__builtin_amdgcn_s_wait_tensorcnt)
#define GAT_USE_TDM 1
#endif
#endif

// ---------------------------- WMMA GEMM ------------------------------------
// C[M,N] = A[M,K] @ B[K,N].  One wave computes one 16-row strip of C across
// TN column tiles (TN accumulators), reusing each A fragment TN times.
// B (K*N floats, row-major, contiguous) is staged into LDS in one TDM DMA.
// Requires: K % 4 == 0, K*N <= 8192, TN*16 >= N.
template <int TN>
__global__ __launch_bounds__(128) void gemm_f32_wmma(
    const float* __restrict__ A, const float* __restrict__ B,
    float* __restrict__ C, int M, int K, int N)
{
    __shared__ float Bs[8192];
    const int tid  = threadIdx.x;
    const int lane = tid & 31;
    const int wave = tid >> 5;

#if GAT_USE_TDM
    if (wave == 0) {
        // Tensor DMA descriptor: 2-D tile (N x K rows), 4-byte elements,
        // contiguous rows (tensor_dim0_stride == N), destination = Bs.
        const unsigned           lds_a = (unsigned)(size_t)&Bs[0];
        const unsigned long long ga    = (unsigned long long)(size_t)B;
        v4u g0 = {0u, 0u, 0u, 0u};
        g0.x = 1u;                                        // count=1, user desc
        g0.y = lds_a;                                     // lds_addr
        g0.z = (unsigned)(ga & 0xFFFFFFFFu);              // global_addr[31:0]
        g0.w = (unsigned)((ga >> 32) & 0x01FFFFFFu)       // global_addr[56:32]
             | 0x80000000u;                               // type = 2 ("image")
        v8i g1 = {0, 0, 0, 0, 0, 0, 0, 0};
        g1[0] = (int)(2u << 16);                          // data_size = 4B
        g1[1] = (int)((unsigned)N << 16);                 // tensor_dim0 lo
        g1[2] = (int)((unsigned)K << 16);                 // tensor_dim1 lo
        g1[3] = (int)((unsigned)N << 16);                 // tile_dim0
        g1[4] = (int)(unsigned)K;                         // tile_dim1
        g1[5] = (int)(unsigned)N;                         // tensor_dim0_stride
        v4i g2 = {0, 0, 0, 0};
        v4i g3 = {0, 0, 0, 0};
        v8i g4 = {0, 0, 0, 0, 0, 0, 0, 0};
        __builtin_amdgcn_tensor_load_to_lds(g0, g1, g2, g3, g4, 0);
        __builtin_amdgcn_s_wait_tensorcnt(0);
    }
#else
    for (int i = tid; i < K * N; i += blockDim.x) Bs[i] = B[i];
#endif
    __syncthreads();

    const int tilesM = (M + 15) >> 4;
    const int tm     = blockIdx.x * 4 + wave;
    if (tm >= tilesM) return;            // wave-uniform: EXEC stays all-1s

    const int half = lane >> 4;          // 0: lanes 0-15, 1: lanes 16-31
    const int lrow = lane & 15;
    int arow = tm * 16 + lrow;
    if (arow >= M) arow = M - 1;         // clamp (avoid OOB, keep EXEC full)

    const float* __restrict__ Arow = A + (size_t)arow * K;

    const v8f zero = {0.f, 0.f, 0.f, 0.f, 0.f, 0.f, 0.f, 0.f};
    v8f acc[TN];
#pragma unroll
    for (int t = 0; t < TN; ++t) acc[t] = zero;

    for (int k = 0; k < K; k += 4) {
        const int kk = k + 2 * half;
        // A fragment: lane holds A[row][kk], A[row][kk+1] (8B-aligned pair)
        const v2f a = *(const v2f*)(Arow + kk);
#pragma unroll
        for (int t = 0; t < TN; ++t) {
            const int col = t * 16 + lrow;
            v2f b;                        // B[kk][col], B[kk+1][col]
            b.x = Bs[kk * N + col];
            b.y = Bs[kk * N + N + col];
            acc[t] = __builtin_amdgcn_wmma_f32_16x16x4_f32(
                false, a, false, b, (short)0, acc[t], false, false);
        }
    }

    // D layout: VGPR r -> M = r (lanes 0-15) / r+8 (lanes 16-31), N = col
    const int rbase = tm * 16 + 8 * half;
#pragma unroll
    for (int t = 0; t < TN; ++t) {
        const int col = t * 16 + lrow;
        if (col < N) {
#pragma unroll
            for (int r = 0; r < 8; ++r) {
                const int rr = rbase + r;
                if (rr < M) C[(size_t)rr * N + col] = acc[t][r];
            }
        }
    }
}

// ----------------------- per-node attention halves -------------------------
// One wave per node: as[n] = h[n,:].a_src ; ad[n] = h[n,:].a_dst
__global__ __launch_bounds__(256) void node_alpha(
    const float* __restrict__ h, const float* __restrict__ a_s,
    const float* __restrict__ a_d, float* __restrict__ as_out,
    float* __restrict__ ad_out, int N, int F)
{
    const int node = blockIdx.x * (blockDim.x >> 5) + (threadIdx.x >> 5);
    const int lane = threadIdx.x & 31;
    if (node >= N) return;
    float vs = 0.f, vd = 0.f;
    const float* __restrict__ row = h + (size_t)node * F;
    for (int f = lane; f < F; f += 32) {
        const float hv = row[f];
        vs += hv * a_s[f];
        vd += hv * a_d[f];
    }
#pragma unroll
    for (int off = 16; off > 0; off >>= 1) {
        vs += __shfl_xor(vs, off, 32);
        vd += __shfl_xor(vd, off, 32);
    }
    if (lane == 0) { as_out[node] = vs; ad_out[node] = vd; }
}

// ---------------------------- helpers --------------------------------------
__global__ void fill_f32(float* __restrict__ p, float v, long long n)
{
    long long i = (long long)blockIdx.x * blockDim.x + threadIdx.x;
    if (i < n) p[i] = v;
}

__device__ __forceinline__ void atomicMaxF32(float* addr, float val)
{
    // sign-split trick: valid for buffers initialized to -inf
    if (val >= 0.f) atomicMax((int*)addr, __float_as_int(val));
    else            atomicMin((unsigned int*)addr, __float_as_uint(val));
}

// ------------------------------ edge passes --------------------------------
// Pass A: e = leaky_relu(as[src] + ad[dst]); segment-max into m[dst]
__global__ __launch_bounds__(256) void edge_score_max(
    const long long* __restrict__ src, const long long* __restrict__ dst,
    long long E, long long Etot, const float* __restrict__ as,
    const float* __restrict__ ad, float* __restrict__ ebuf,
    float* __restrict__ m)
{
    long long t = (long long)blockIdx.x * blockDim.x + threadIdx.x;
    if (t >= Etot) return;
    long long s, d;
    if (t < E) { s = src[t]; d = dst[t]; } else { s = t - E; d = s; }
    float v = as[s] + ad[d];
    v = (v > 0.f) ? v : 0.2f * v;        // leaky_relu, slope 0.2
    ebuf[t] = v;
    atomicMaxF32(m + d, v);
}

// Pass B: ex = exp(e - m[dst]); segment-sum into den[dst]; ebuf <- ex
__global__ __launch_bounds__(256) void edge_exp_sum(
    const long long* __restrict__ src, const long long* __restrict__ dst,
    long long E, long long Etot, const float* __restrict__ m,
    float* __restrict__ ebuf, float* __restrict__ den)
{
    long long t = (long long)blockIdx.x * blockDim.x + threadIdx.x;
    if (t >= Etot) return;
    long long d = (t < E) ? dst[t] : (t - E);
    float ex = __expf(ebuf[t] - m[d]);
    ebuf[t] = ex;
    atomicAdd(den + d, ex);
}

// Pass C: out[dst,:] += h[src,:] * (ex/den[dst]).  One wave per edge.
__global__ __launch_bounds__(256) void edge_aggregate(
    const long long* __restrict__ src, const long long* __restrict__ dst,
    long long E, long long Etot, const float* __restrict__ ebuf,
    const float* __restrict__ den, const float* __restrict__ h,
    float* __restrict__ out, int F)
{
    long long w = (long long)blockIdx.x * (blockDim.x >> 5) + (threadIdx.x >> 5);
    const int lane = threadIdx.x & 31;
    if (w >= Etot) return;
    long long s, d;
    if (w < E) { s = src[w]; d = dst[w]; } else { s = w - E; d = s; }
    const float coef = ebuf[w] / den[d];
    const float* __restrict__ hs = h + (size_t)s * F;
    float* __restrict__ o = out + (size_t)d * F;
    for (int f = lane; f < F; f += 32)
        atomicAdd(o + f, hs[f] * coef);
}

// out[i,f] = relu(out[i,f] + b[f])
__global__ __launch_bounds__(256) void bias_relu(
    float* __restrict__ out, const float* __restrict__ b, long long n, int F)
{
    long long i = (long long)blockIdx.x * blockDim.x + threadIdx.x;
    if (i >= n) return;
    float v = out[i] + b[i % F];
    out[i] = v > 0.f ? v : 0.f;
}

// row <- log_softmax(row + b).  One wave per node.
__global__ __launch_bounds__(256) void bias_logsoftmax(
    float* __restrict__ out, const float* __restrict__ b, int N, int F)
{
    const int node = blockIdx.x * (blockDim.x >> 5) + (threadIdx.x >> 5);
    const int lane = threadIdx.x & 31;
    if (node >= N) return;
    float* __restrict__ row = out + (size_t)node * F;

    float mx = -INFINITY;
    for (int f = lane; f < F; f += 32) mx = fmaxf(mx, row[f] + b[f]);
#pragma unroll
    for (int off = 16; off > 0; off >>= 1) mx = fmaxf(mx, __shfl_xor(mx, off, 32));

    float s = 0.f;
    for (int f = lane; f < F; f += 32) s += __expf(row[f] + b[f] - mx);
#pragma unroll
    for (int off = 16; off > 0; off >>= 1) s += __shfl_xor(s, off, 32);
    const float lse = logf(s);

    for (int f = lane; f < F; f += 32) row[f] = row[f] + b[f] - mx - lse;
}

// ---------------------------------------------------------------------------
extern "C" void kernel_launch(void* const* d_in, const int* in_sizes, int n_in,
                              void* d_out, int out_size, void* d_ws, size_t ws_size,
                              hipStream_t stream)
{
    const float*     x    = (const float*)d_in[0];
    const long long* ei   = (const long long*)d_in[1];   // int64 [2, E]
    const float*     W1   = (const float*)d_in[2];
    const float*     asr1 = (const float*)d_in[3];
    const float*     adt1 = (const float*)d_in[4];
    const float*     b1   = (const float*)d_in[5];
    const float*     W2   = (const float*)d_in[6];
    const float*     asr2 = (const float*)d_in[7];
    const float*     adt2 = (const float*)d_in[8];
    const float*     b2   = (const float*)d_in[9];

    const int Fin  = 128;
    const int N    = in_sizes[0] / Fin;          // 100000
    const long long E = in_sizes[1] / 2;         // 1.6M
    const int Fh   = in_sizes[3];                // 64
    const int Fo   = in_sizes[7];                // 40
    const long long Etot = E + (long long)N;     // + self loops

    const long long* src = ei;
    const long long* dst = ei + E;

    // workspace layout (floats)
    float* ws   = (float*)d_ws;
    float* h1   = ws;                               // N*Fh
    float* agg1 = h1   + (size_t)N * Fh;            // N*Fh
    float* h2   = agg1 + (size_t)N * Fh;            // N*Fo
    float* ebuf = h2   + (size_t)N * Fo;            // Etot
    float* as1  = ebuf + Etot;                      // N
    float* ad1  = as1 + N;
    float* m1   = ad1 + N;
    float* den1 = m1  + N;
    float* as2  = den1 + N;
    float* ad2  = as2 + N;
    float* m2   = ad2 + N;
    float* den2 = m2  + N;

    float* out  = (float*)d_out;                    // N*Fo

    const int tilesM = (N + 15) / 16;               // 6250
    auto cdiv = [](long long a, long long b) { return (int)((a + b - 1) / b); };

    // ---------------- layer 1 ----------------
    // h1 = x @ W1   (K=128, N=64 -> 4 column tiles per wave)
    gemm_f32_wmma<4><<<cdiv(tilesM, 4), 128, 0, stream>>>(x, W1, h1, N, Fin, Fh);

    node_alpha<<<cdiv(N, 8), 256, 0, stream>>>(h1, asr1, adt1, as1, ad1, N, Fh);

    fill_f32<<<cdiv(N, 256), 256, 0, stream>>>(m1,  -INFINITY, N);
    fill_f32<<<cdiv(N, 256), 256, 0, stream>>>(den1, 0.f,      N);
    fill_f32<<<cdiv((long long)N * Fh, 256), 256, 0, stream>>>(agg1, 0.f, (long long)N * Fh);

    edge_score_max<<<cdiv(Etot, 256), 256, 0, stream>>>(src, dst, E, Etot, as1, ad1, ebuf, m1);
    edge_exp_sum  <<<cdiv(Etot, 256), 256, 0, stream>>>(src, dst, E, Etot, m1, ebuf, den1);
    edge_aggregate<<<cdiv(Etot, 8),   256, 0, stream>>>(src, dst, E, Etot, ebuf, den1, h1, agg1, Fh);
    bias_relu<<<cdiv((long long)N * Fh, 256), 256, 0, stream>>>(agg1, b1, (long long)N * Fh, Fh);

    // ---------------- layer 2 ----------------
    // h2 = agg1 @ W2   (K=64, N=40 -> 3 column tiles per wave)
    gemm_f32_wmma<3><<<cdiv(tilesM, 4), 128, 0, stream>>>(agg1, W2, h2, N, Fh, Fo);

    node_alpha<<<cdiv(N, 8), 256, 0, stream>>>(h2, asr2, adt2, as2, ad2, N, Fo);

    fill_f32<<<cdiv(N, 256), 256, 0, stream>>>(m2,  -INFINITY, N);
    fill_f32<<<cdiv(N, 256), 256, 0, stream>>>(den2, 0.f,      N);
    fill_f32<<<cdiv((long long)N * Fo, 256), 256, 0, stream>>>(out, 0.f, (long long)N * Fo);

    edge_score_max<<<cdiv(Etot, 256), 256, 0, stream>>>(src, dst, E, Etot, as2, ad2, ebuf, m2);
    edge_exp_sum  <<<cdiv(Etot, 256), 256, 0, stream>>>(src, dst, E, Etot, m2, ebuf, den2);
    edge_aggregate<<<cdiv(Etot, 8),   256, 0, stream>>>(src, dst, E, Etot, ebuf, den2, h2, out, Fo);

    bias_logsoftmax<<<cdiv(N, 8), 256, 0, stream>>>(out, b2, N, Fo);
}